// HGNN_89361089560892
// MI455X (gfx1250) — compile-verified
//
#include <hip/hip_runtime.h>
#include <hip/hip_bf16.h>

typedef __attribute__((ext_vector_type(2))) float v2f;
typedef __attribute__((ext_vector_type(8))) float v8f;

#define NEG_SLOPE 0.2f

// D = A(16x4 f32) x B(4x16 f32) + C(16x16 f32), wave32 WMMA
__device__ __forceinline__ v8f wmma4(v2f a, v2f b, v8f c) {
  return __builtin_amdgcn_wmma_f32_16x16x4_f32(false, a, false, b, (short)0, c,
                                               false, false);
}

__device__ __forceinline__ float leaky(float x) {
  return x > 0.0f ? x : NEG_SLOPE * x;
}

__device__ __forceinline__ float vsum8(v8f c) {
  return ((c[0] + c[1]) + (c[2] + c[3])) + ((c[4] + c[5]) + (c[6] + c[7]));
}

// ---------------------------------------------------------------------------
// Kernel 1 (dsd level 2): emb_s1[B*H1, 64] =
//   leaky( es@W21^T + mean_h2( ed@W21^T + (ed*es)@W22^T ) )
// Rows = B*H1*H2 (65536); group = 8 rows (h2); 16-row tile = 2 groups.
// Block = 128 threads = 4 waves, wave w -> column tile n0 = 16*w.
// ---------------------------------------------------------------------------
__global__ void k_dsd1(const float* __restrict__ Es, const float* __restrict__ Ed,
                       const float* __restrict__ W21, const float* __restrict__ W22,
                       const int* __restrict__ dsd1, const int* __restrict__ dsd2,
                       float* __restrict__ emb_s1) {
  const int lane  = threadIdx.x & 31;
  const int n0    = (threadIdx.x >> 5) << 4;
  const int r0    = blockIdx.x << 4;
  const int m     = lane & 15;
  const int r     = r0 + m;            // gathered neighbor row
  const int g     = r >> 3;            // (b,h1) group of my A row
  const int khalf = (lane >> 4) << 1;  // 0 for lanes 0-15, 2 for 16-31

  const float* rowE = Ed + (size_t)dsd2[r] * 64;
  const float* rowS = Es + (size_t)dsd1[g] * 64;

  int cnt = 0;
  const int* gi = dsd2 + ((size_t)g << 3);
#pragma unroll
  for (int j = 0; j < 8; ++j) cnt += (gi[j] != 0) ? 1 : 0;
  const float w = (cnt > 0) ? 1.0f / ((float)cnt + 1e-8f) : 0.0f;

  v8f c = {};
#pragma unroll
  for (int k0 = 0; k0 < 64; k0 += 4) {
    const int k = k0 + khalf;
    v2f ae = *(const v2f*)(rowE + k);
    v2f as = *(const v2f*)(rowS + k);
    v2f a1 = ae * w;        // pre-scaled by 1/cnt -> VGPR-sum == masked mean
    v2f a2 = ae * as * w;
    const size_t wo = (size_t)(n0 + m) * 64 + k;   // B[k][n] = W[n][k]
    v2f b1 = *(const v2f*)(W21 + wo);
    v2f b2 = *(const v2f*)(W22 + wo);
    c = wmma4(a1, b1, c);
    c = wmma4(a2, b2, c);
  }
  const float s = vsum8(c);  // sum over my half's 8 rows == mean over h2

  // self message for output element (og, n0+m)
  const int og = (r0 >> 3) + (lane >> 4);
  const float* rowSo = Es + (size_t)dsd1[og] * 64;
  const float* wr    = W21 + (size_t)(n0 + m) * 64;
  float dot = 0.0f;
#pragma unroll
  for (int k = 0; k < 64; ++k) dot += rowSo[k] * wr[k];
  emb_s1[(size_t)og * 64 + n0 + m] = leaky(dot + s);
}

// ---------------------------------------------------------------------------
// Kernel 2 (dsd level 1): emb_dise[B, 64] =
//   leaky( td@W11^T + mean_h1( emb_s1@W11^T + (emb_s1*td)@W12^T ) )
// Rows = B*H1 (8192); group = 8 rows (h1); tile = 2 batches.
// ---------------------------------------------------------------------------
__global__ void k_dsd2(const float* __restrict__ Ed, const float* __restrict__ X,
                       const float* __restrict__ W11, const float* __restrict__ W12,
                       const int* __restrict__ label, const int* __restrict__ dsd1,
                       float* __restrict__ emb_dise) {
  const int lane  = threadIdx.x & 31;
  const int n0    = (threadIdx.x >> 5) << 4;
  const int r0    = blockIdx.x << 4;
  const int m     = lane & 15;
  const int r     = r0 + m;
  const int b     = r >> 3;
  const int khalf = (lane >> 4) << 1;

  const float* rowX = X + (size_t)r * 64;
  const float* rowT = Ed + (size_t)label[b] * 64;

  int cnt = 0;
  const int* gi = dsd1 + ((size_t)b << 3);
#pragma unroll
  for (int j = 0; j < 8; ++j) cnt += (gi[j] != 0) ? 1 : 0;
  const float w = (cnt > 0) ? 1.0f / ((float)cnt + 1e-8f) : 0.0f;

  v8f c = {};
#pragma unroll
  for (int k0 = 0; k0 < 64; k0 += 4) {
    const int k = k0 + khalf;
    v2f x = *(const v2f*)(rowX + k);
    v2f t = *(const v2f*)(rowT + k);
    v2f a1 = x * w;
    v2f a2 = x * t * w;
    const size_t wo = (size_t)(n0 + m) * 64 + k;
    v2f b1 = *(const v2f*)(W11 + wo);
    v2f b2 = *(const v2f*)(W12 + wo);
    c = wmma4(a1, b1, c);
    c = wmma4(a2, b2, c);
  }
  const float s = vsum8(c);

  const int ob = (r0 >> 3) + (lane >> 4);
  const float* rowTo = Ed + (size_t)label[ob] * 64;
  const float* wr    = W11 + (size_t)(n0 + m) * 64;
  float dot = 0.0f;
#pragma unroll
  for (int k = 0; k < 64; ++k) dot += rowTo[k] * wr[k];
  emb_dise[(size_t)ob * 64 + n0 + m] = leaky(dot + s);
}

// ---------------------------------------------------------------------------
// Kernel 3 (usu level 3): eu2[B*U1*U2, 64] = leaky( mean_u3( s3@W3^T ) )
// Rows = B*U1*U2*U3 (131072); group = 16 rows = whole tile; combine lane
// halves with a 16-lane xor shuffle.
// ---------------------------------------------------------------------------
__global__ void k_usu1(const float* __restrict__ Es, const float* __restrict__ W3,
                       const int* __restrict__ usu3, float* __restrict__ eu2) {
  const int lane  = threadIdx.x & 31;
  const int n0    = (threadIdx.x >> 5) << 4;
  const int r0    = blockIdx.x << 4;
  const int m     = lane & 15;
  const int r     = r0 + m;
  const int khalf = (lane >> 4) << 1;

  const float* rowX = Es + (size_t)usu3[r] * 64;

  int cnt = 0;
  const int* gi = usu3 + r0;
#pragma unroll
  for (int j = 0; j < 16; ++j) cnt += (gi[j] != 0) ? 1 : 0;
  const float w = (cnt > 0) ? 1.0f / ((float)cnt + 1e-8f) : 0.0f;

  v8f c = {};
#pragma unroll
  for (int k0 = 0; k0 < 64; k0 += 4) {
    const int k = k0 + khalf;
    v2f a = *(const v2f*)(rowX + k) * w;
    v2f b = *(const v2f*)(W3 + (size_t)(n0 + m) * 64 + k);
    c = wmma4(a, b, c);
  }
  float s = vsum8(c);           // sum over 8 rows of my half
  s += __shfl_xor(s, 16, 32);   // + other half -> sum over all 16 (u3)
  if (lane < 16) eu2[(size_t)(r0 >> 4) * 64 + n0 + lane] = leaky(s);
}

// ---------------------------------------------------------------------------
// Kernel 4 (usu level 2): es1[B*U1, 64] =
//   leaky( u1@W21u^T + mean_u2( eu2@W21u^T + (eu2*u1)@W22u^T ) )
// ---------------------------------------------------------------------------
__global__ void k_usu2(const float* __restrict__ Es, const float* __restrict__ X,
                       const float* __restrict__ W21, const float* __restrict__ W22,
                       const int* __restrict__ usu1, const int* __restrict__ usu2,
                       float* __restrict__ es1) {
  const int lane  = threadIdx.x & 31;
  const int n0    = (threadIdx.x >> 5) << 4;
  const int r0    = blockIdx.x << 4;
  const int m     = lane & 15;
  const int r     = r0 + m;
  const int g     = r >> 3;    // (b,u1)
  const int khalf = (lane >> 4) << 1;

  const float* rowX = X + (size_t)r * 64;
  const float* rowU = Es + (size_t)usu1[g] * 64;

  int cnt = 0;
  const int* gi = usu2 + ((size_t)g << 3);
#pragma unroll
  for (int j = 0; j < 8; ++j) cnt += (gi[j] != 0) ? 1 : 0;
  const float w = (cnt > 0) ? 1.0f / ((float)cnt + 1e-8f) : 0.0f;

  v8f c = {};
#pragma unroll
  for (int k0 = 0; k0 < 64; k0 += 4) {
    const int k = k0 + khalf;
    v2f x = *(const v2f*)(rowX + k);
    v2f u = *(const v2f*)(rowU + k);
    v2f a1 = x * w;
    v2f a2 = x * u * w;
    const size_t wo = (size_t)(n0 + m) * 64 + k;
    v2f b1 = *(const v2f*)(W21 + wo);
    v2f b2 = *(const v2f*)(W22 + wo);
    c = wmma4(a1, b1, c);
    c = wmma4(a2, b2, c);
  }
  const float s = vsum8(c);

  const int og = (r0 >> 3) + (lane >> 4);
  const float* rowUo = Es + (size_t)usu1[og] * 64;
  const float* wr    = W21 + (size_t)(n0 + m) * 64;
  float dot = 0.0f;
#pragma unroll
  for (int k = 0; k < 64; ++k) dot += rowUo[k] * wr[k];
  es1[(size_t)og * 64 + n0 + m] = leaky(dot + s);
}

// ---------------------------------------------------------------------------
// Kernel 5 (usu level 1): emb_user[B, 64] = leaky( mean_u1( es1@W1u^T ) )
// ---------------------------------------------------------------------------
__global__ void k_usu3(const float* __restrict__ X, const float* __restrict__ W1,
                       const int* __restrict__ usu1, float* __restrict__ emb_user) {
  const int lane  = threadIdx.x & 31;
  const int n0    = (threadIdx.x >> 5) << 4;
  const int r0    = blockIdx.x << 4;
  const int m     = lane & 15;
  const int r     = r0 + m;
  const int b     = r >> 3;
  const int khalf = (lane >> 4) << 1;

  const float* rowX = X + (size_t)r * 64;

  int cnt = 0;
  const int* gi = usu1 + ((size_t)b << 3);
#pragma unroll
  for (int j = 0; j < 8; ++j) cnt += (gi[j] != 0) ? 1 : 0;
  const float w = (cnt > 0) ? 1.0f / ((float)cnt + 1e-8f) : 0.0f;

  v8f c = {};
#pragma unroll
  for (int k0 = 0; k0 < 64; k0 += 4) {
    const int k = k0 + khalf;
    v2f a = *(const v2f*)(rowX + k) * w;
    v2f bb = *(const v2f*)(W1 + (size_t)(n0 + m) * 64 + k);
    c = wmma4(a, bb, c);
  }
  const float s = vsum8(c);

  const int ob = (r0 >> 3) + (lane >> 4);
  emb_user[(size_t)ob * 64 + n0 + m] = leaky(s);
}

// ---------------------------------------------------------------------------
// Kernel 6: pred[b] = dot(emb_dise[b], emb_user[b]); block = 64 threads.
// ---------------------------------------------------------------------------
__global__ void k_pred(const float* __restrict__ emb_dise,
                       const float* __restrict__ emb_user,
                       float* __restrict__ out) {
  const int b = blockIdx.x;
  const int t = threadIdx.x;
  __shared__ float red[64];
  red[t] = emb_dise[(size_t)b * 64 + t] * emb_user[(size_t)b * 64 + t];
  __syncthreads();
#pragma unroll
  for (int off = 32; off > 0; off >>= 1) {
    if (t < off) red[t] += red[t + off];
    __syncthreads();
  }
  if (t == 0) out[b] = red[0];
}

extern "C" void kernel_launch(void* const* d_in, const int* in_sizes, int n_in,
                              void* d_out, int out_size, void* d_ws, size_t ws_size,
                              hipStream_t stream) {
  (void)in_sizes; (void)n_in; (void)out_size; (void)ws_size;
  const float* Es       = (const float*)d_in[0];
  const float* Ed       = (const float*)d_in[1];
  const float* W_dsd_21 = (const float*)d_in[2];
  const float* W_dsd_22 = (const float*)d_in[3];
  const float* W_dsd_11 = (const float*)d_in[4];
  const float* W_dsd_12 = (const float*)d_in[5];
  const float* W_usu_3  = (const float*)d_in[6];
  const float* W_usu_21 = (const float*)d_in[7];
  const float* W_usu_22 = (const float*)d_in[8];
  const float* W_usu_1  = (const float*)d_in[9];
  const int* label = (const int*)d_in[10];
  const int* dsd1  = (const int*)d_in[11];
  const int* dsd2  = (const int*)d_in[12];
  const int* usu1  = (const int*)d_in[13];
  const int* usu2  = (const int*)d_in[14];
  const int* usu3  = (const int*)d_in[15];

  float* ws       = (float*)d_ws;
  float* emb_s1   = ws;                       // 8192*64
  float* eu2      = emb_s1 + 8192 * 64;       // 65536*64
  float* es1      = eu2 + 65536 * 64;         // 8192*64
  float* emb_dise = es1 + 8192 * 64;          // 1024*64
  float* emb_user = emb_dise + 1024 * 64;     // 1024*64
  float* out      = (float*)d_out;

  const dim3 blk(128);
  k_dsd1<<<4096, blk, 0, stream>>>(Es, Ed, W_dsd_21, W_dsd_22, dsd1, dsd2, emb_s1);
  k_dsd2<<<512,  blk, 0, stream>>>(Ed, emb_s1, W_dsd_11, W_dsd_12, label, dsd1, emb_dise);
  k_usu1<<<8192, blk, 0, stream>>>(Es, W_usu_3, usu3, eu2);
  k_usu2<<<4096, blk, 0, stream>>>(Es, eu2, W_usu_21, W_usu_22, usu1, usu2, es1);
  k_usu3<<<512,  blk, 0, stream>>>(es1, W_usu_1, usu1, emb_user);
  k_pred<<<1024, dim3(64), 0, stream>>>(emb_dise, emb_user, out);
}